// GreedyFeatureInit_60790967107716
// MI455X (gfx1250) — compile-verified
//
#include <hip/hip_runtime.h>

// GreedyFeatureInit for MI455X (gfx1250, wave32)
// B=64, N=1024 (32x32 grid), D=768, 11 slots, r_avg=1, r_supp=2, strength=0.5

#define Bsz   64
#define Nn    1024
#define Dd    768
#define HW    32
#define SLOTS 11

typedef __attribute__((ext_vector_type(2))) float v2f;
typedef __attribute__((ext_vector_type(8))) float v8f;

// ---------------------------------------------------------------------------
// Kernel 1: saliency^2[b,n] = sum_d f[b,n,d]^2   (memory-bound: 192 MB read)
// One wave32 per row; float4 loads; wave shuffle reduction.
// ---------------------------------------------------------------------------
__global__ __launch_bounds__(256) void sal2_kernel(const float* __restrict__ feat,
                                                   float* __restrict__ sal2) {
    const int row  = blockIdx.x * 8 + (threadIdx.x >> 5);   // 8 waves / block
    const int lane = threadIdx.x & 31;
    const float4* rp = (const float4*)(feat + (size_t)row * Dd);  // 192 float4 per row
    float acc = 0.0f;
#pragma unroll
    for (int j = 0; j < 6; ++j) {
        float4 v = rp[lane + 32 * j];
        acc = fmaf(v.x, v.x, fmaf(v.y, v.y, fmaf(v.z, v.z, fmaf(v.w, v.w, acc))));
    }
#pragma unroll
    for (int off = 16; off; off >>= 1) acc += __shfl_xor(acc, off, 32);
    if (lane == 0) sal2[row] = acc;
}

// ---------------------------------------------------------------------------
// Kernel 2: greedy loop. One workgroup (4 waves / 128 thr) per batch element.
// Scores (squared domain) live in LDS. Neighbor pooling done with
// v_wmma_f32_16x16x4_f32 (A = all-ones -> D[m,n] = sum_k B[k,n], layout-robust).
// ---------------------------------------------------------------------------
__global__ __launch_bounds__(128) void greedy_kernel(const float* __restrict__ feat,
                                                     const float* __restrict__ sal2_in,
                                                     float* __restrict__ out) {
    __shared__ float sal[Nn];
    __shared__ float rv[128];
    __shared__ int   ri[128];
    __shared__ int   selIdx;

    const int b    = blockIdx.x;
    const int tid  = threadIdx.x;
    const int lane = tid & 31;
    const int wave = tid >> 5;
    const float* fb = feat + (size_t)b * Nn * Dd;

    for (int i = tid; i < Nn; i += 128) sal[i] = sal2_in[b * Nn + i];
    __syncthreads();

    for (int s = 0; s < SLOTS; ++s) {
        // ---- argmax, first-occurrence tie-break (contiguous per-thread ranges) ----
        float best = -__builtin_inff();
        int bidx = tid * 8;
        for (int k = 0; k < 8; ++k) {
            int i = tid * 8 + k;
            float v = sal[i];
            if (v > best) { best = v; bidx = i; }
        }
        rv[tid] = best; ri[tid] = bidx;
        __syncthreads();
        if (tid == 0) {
            float bb = rv[0]; int bi = ri[0];
            for (int t = 1; t < 128; ++t)
                if (rv[t] > bb) { bb = rv[t]; bi = ri[t]; }
            selIdx = bi;
        }
        __syncthreads();
        const int idx = selIdx;
        const int h = idx >> 5, w = idx & 31;

        // ---- neighbor list: 3x3 box clipped at borders, padded to 16 slots ----
        int plist[16]; int cnt = 0;
        for (int dy = -1; dy <= 1; ++dy)
            for (int dx = -1; dx <= 1; ++dx) {
                int hh = h + dy, ww = w + dx;
                if (hh >= 0 && hh < HW && ww >= 0 && ww < HW) plist[cnt++] = hh * HW + ww;
            }
        const float inv = 1.0f / (float)cnt;
        for (int i = cnt; i < 16; ++i) plist[i] = -1;

        // ---- WMMA pooling: column-sum of 16 padded neighbor rows, per 16-dim chunk ----
        const v2f onesA = {1.0f, 1.0f};        // A == all ones -> D = colsum(B) over K
        const int half = lane >> 4;            // which half-wave (B lane-half slot)
        const int n    = lane & 15;            // N = lane % 16 (column / dim-in-chunk)
        for (int c = wave; c < Dd / 16; c += 4) {   // 48 chunks over 4 waves, uniform
            const int d = c * 16 + n;
            v8f acc = {};
#pragma unroll
            for (int g = 0; g < 4; ++g) {
                int s0 = 4 * g + 2 * half;
                int s1 = s0 + 1;
                int p0 = plist[s0], p1 = plist[s1];
                int q0 = p0 < 0 ? 0 : p0;
                int q1 = p1 < 0 ? 0 : p1;
                float x0 = fb[(size_t)q0 * Dd + d];
                float x1 = fb[(size_t)q1 * Dd + d];
                v2f bm;
                bm.x = (p0 < 0) ? 0.0f : x0;
                bm.y = (p1 < 0) ? 0.0f : x1;
                // D = A(16x4 ones) * B(4x16 neighbor chunk) + C
                acc = __builtin_amdgcn_wmma_f32_16x16x4_f32(
                    false, onesA, false, bm, (short)0, acc, false, false);
            }
            // all D rows identical; row M=0 lives in acc[0], N = lane (lanes 0..15)
            if (lane < 16)
                out[((size_t)b * SLOTS + s) * Dd + d] = acc[0] * inv;
        }

        // ---- suppression in squared domain: sal2 *= g^2 ; winner -> -inf ----
        for (int i = tid; i < Nn; i += 128) {
            float dh = fabsf((float)(i >> 5) - (float)h);
            float dw = fabsf((float)(i & 31) - (float)w);
            float dist = fmaxf(dh, dw);
            float supp = fminf(dist * (1.0f / 3.0f), 1.0f);
            float g = 1.0f - 0.5f * (1.0f - supp);
            sal[i] = (i == idx) ? -__builtin_inff() : sal[i] * (g * g);
        }
        __syncthreads();
    }
}

// ---------------------------------------------------------------------------
extern "C" void kernel_launch(void* const* d_in, const int* in_sizes, int n_in,
                              void* d_out, int out_size, void* d_ws, size_t ws_size,
                              hipStream_t stream) {
    const float* feat = (const float*)d_in[0];       // [B, N, D] f32
    float* out  = (float*)d_out;                     // [B, SLOTS, D] f32
    float* sal2 = (float*)d_ws;                      // B*N floats = 256 KB scratch

    sal2_kernel<<<(Bsz * Nn) / 8, 256, 0, stream>>>(feat, sal2);
    greedy_kernel<<<Bsz, 128, 0, stream>>>(feat, sal2, out);
}